// GCNLayer_63900523430084
// MI455X (gfx1250) — compile-verified
//
#include <hip/hip_runtime.h>

typedef __attribute__((ext_vector_type(2))) float v2f;
typedef __attribute__((ext_vector_type(8))) float v8f;

// ---------------------------------------------------------------------------
// Kernel 1: zero the output accumulator (harness poisons d_out with 0xAA).
// ---------------------------------------------------------------------------
__global__ void spmm_zero_kernel(float* __restrict__ out, int n) {
  int i = blockIdx.x * blockDim.x + threadIdx.x;
  int stride = gridDim.x * blockDim.x;
  for (; i < n; i += stride) out[i] = 0.0f;
}

// ---------------------------------------------------------------------------
// SpMM tile: one wave (32 lanes) handles 16 edges x 32 features.
//   D(16x16) = sum_kb A_kb(16x4) x B_kb(4x16),  A_kb = diag(vals) block,
//   B_kb = gathered embeds rows. Two chains (feature halves) = 8 WMMAs/wave.
// FULL=true: whole tile in range -> unguarded loads + branch-free atomics.
// All global accesses use uniform-SGPR base + signed 32-bit per-lane index
// so the backend emits the GVS (saddr + voffset scale_offset) form.
// ---------------------------------------------------------------------------
template <bool FULL>
__device__ __forceinline__ void spmm_tile(
    const int* __restrict__ rows, const int* __restrict__ cols,
    const float* __restrict__ vals, const float* __restrict__ embeds,
    float* __restrict__ out, int E, int edge_base, int lane) {
  const int n  = lane & 15;   // N index (feature within 16-wide tile) == M row
  const int hi = lane >> 4;   // upper/lower half of wave

  // A-matrix source: lane L holds vals for edge m = L%16 (diagonal entry).
  float vA;
  if (FULL) {
    vA = vals[edge_base + n];
  } else {
    int e = edge_base + n;
    vA = (e < E) ? vals[e] : 0.0f;
  }

  // Build diagonal A blocks once (independent of feature half) and gather
  // column indices for the B rows this lane supplies.
  // A 16x4 layout: lane = M + 16*(K>=2), vgpr = K%2.
  // B 4x16 layout: vgpr0 -> K = 2*hi, vgpr1 -> K = 2*hi+1, N = lane%16.
  float a0[4], a1[4];
  int   b0[4], b1[4];   // element indices into embeds (uniform base + idx)
#pragma unroll
  for (int kb = 0; kb < 4; ++kb) {
    const int k0 = kb * 4 + 2 * hi;      // global K of this lane's .x slot
    a0[kb] = (n == k0)     ? vA : 0.0f;
    a1[kb] = (n == k0 + 1) ? vA : 0.0f;
    const int e0 = edge_base + k0;
    int c0, c1;
    if (FULL) {
      c0 = cols[e0];
      c1 = cols[e0 + 1];
    } else {
      c0 = (e0 < E)     ? cols[e0]     : 0;
      c1 = (e0 + 1 < E) ? cols[e0 + 1] : 0;
    }
    b0[kb] = c0 * 32 + n;
    b1[kb] = c1 * 32 + n;
  }

  // Destination element indices for the scatter (fold the lane's feature n
  // in now so the atomic is out[uniform base + per-lane int index]).
  // D layout: vgpr v -> M = v + 8*hi.
  int didx[8];
#pragma unroll
  for (int v = 0; v < 8; ++v) {
    const int e = edge_base + v + 8 * hi;
    const int r = FULL ? rows[e] : ((e < E) ? rows[e] : 0);
    didx[v] = r * 32 + n;
  }

#pragma unroll
  for (int h = 0; h < 2; ++h) {          // two 16-wide feature halves (D=32)
    v8f acc = {};
#pragma unroll
    for (int kb = 0; kb < 4; ++kb) {
      v2f A, B;
      A.x = a0[kb];
      A.y = a1[kb];
      // 16 consecutive floats of a gathered row per half-wave (coalesced);
      // +16*h folds into the instruction immediate offset (64 bytes).
      B.x = embeds[b0[kb] + 16 * h];
      B.y = embeds[b1[kb] + 16 * h];
      acc = __builtin_amdgcn_wmma_f32_16x16x4_f32(
          /*neg_a=*/false, A, /*neg_b=*/false, B,
          /*c_mod=*/(short)0, acc, /*reuse_a=*/false, /*reuse_b=*/false);
    }
    // Scatter: D[m, n] -> out[rows[edge_base+m]*32 + 16*h + n], m = v + 8*hi.
#pragma unroll
    for (int v = 0; v < 8; ++v) {
      if (FULL || (edge_base + v + 8 * hi) < E) {
        __hip_atomic_fetch_add(out + (didx[v] + 16 * h), acc[v],
                               __ATOMIC_RELAXED, __HIP_MEMORY_SCOPE_AGENT);
      }
    }
  }
}

__global__ __launch_bounds__(256) void spmm_wmma_kernel(
    const int* __restrict__ rows, const int* __restrict__ cols,
    const float* __restrict__ vals, const float* __restrict__ embeds,
    float* __restrict__ out, int E) {
  const int lane = threadIdx.x & 31;
  const int wave = threadIdx.x >> 5;
  const int edge_base = (blockIdx.x * 8 + wave) * 16;
  if (edge_base >= E) return;   // wave-uniform: EXEC stays all-1s for WMMA

  // Prefetch the streaming edge arrays ~8KB ahead (speculative, OOB dropped).
  __builtin_prefetch(cols + edge_base + 2048, 0, 1);
  __builtin_prefetch(rows + edge_base + 2048, 0, 1);
  __builtin_prefetch(vals + edge_base + 2048, 0, 1);

  if (edge_base + 16 <= E) {    // wave-uniform fast/tail split
    spmm_tile<true>(rows, cols, vals, embeds, out, E, edge_base, lane);
  } else {
    spmm_tile<false>(rows, cols, vals, embeds, out, E, edge_base, lane);
  }
}

// ---------------------------------------------------------------------------
// Launch: inputs are (rows i32[E], cols i32[E], vals f32[E], embeds f32[N*32]).
// Output: f32[N*32].
// ---------------------------------------------------------------------------
extern "C" void kernel_launch(void* const* d_in, const int* in_sizes, int n_in,
                              void* d_out, int out_size, void* d_ws, size_t ws_size,
                              hipStream_t stream) {
  const int*   rows   = (const int*)d_in[0];
  const int*   cols   = (const int*)d_in[1];
  const float* vals   = (const float*)d_in[2];
  const float* embeds = (const float*)d_in[3];
  float*       out    = (float*)d_out;
  const int E = in_sizes[0];

  // Zero the accumulator.
  int zblocks = (out_size + 255) / 256;
  if (zblocks > 16384) zblocks = 16384;  // grid-stride covers the rest
  spmm_zero_kernel<<<zblocks, 256, 0, stream>>>(out, out_size);

  // 8 waves per block, 16 edges per wave -> 128 edges per block.
  int blocks = (E + 127) / 128;
  spmm_wmma_kernel<<<blocks, 256, 0, stream>>>(rows, cols, vals, embeds, out, E);
}